// WeightedGCNLayer_7851200217799
// MI455X (gfx1250) — compile-verified
//
#include <hip/hip_runtime.h>

typedef __attribute__((ext_vector_type(2))) float v2f;
typedef __attribute__((ext_vector_type(8))) float v8f;

#define D 128
#define LDS_STRIDE 132  // +4 floats pad: conflict-free ds_load_b64 fragment reads

// ---------------- Phase 0: zero the aggregation buffer ----------------
__global__ __launch_bounds__(256) void gcn_zero_kernel(float4* __restrict__ agg4,
                                                       int total4) {
  int t = blockIdx.x * blockDim.x + threadIdx.x;
  if (t < total4) agg4[t] = make_float4(0.f, 0.f, 0.f, 0.f);
}

// ---------------- Phase 1: edge scatter (one wave32 per edge) ----------------
// lane L owns floats [4L, 4L+4) of the 128-float feature row.
__global__ __launch_bounds__(256) void gcn_scatter_kernel(
    const float* __restrict__ node_emb, const float* __restrict__ edge_weight,
    const int* __restrict__ src, const int* __restrict__ dst,
    float* __restrict__ agg, int E) {
  int eid = (int)((blockIdx.x * (unsigned)blockDim.x + threadIdx.x) >> 5);
  int lane = threadIdx.x & 31;
  if (eid >= E) return;
  int s = src[eid];
  int d = dst[eid];
  float w = edge_weight[eid];
  const float4 v = *(const float4*)(node_emb + (size_t)s * D + lane * 4);
  float* p = agg + (size_t)d * D + lane * 4;
  // relaxed agent-scope fp32 atomics -> non-returning global_atomic_add_f32
  __hip_atomic_fetch_add(p + 0, v.x * w, __ATOMIC_RELAXED, __HIP_MEMORY_SCOPE_AGENT);
  __hip_atomic_fetch_add(p + 1, v.y * w, __ATOMIC_RELAXED, __HIP_MEMORY_SCOPE_AGENT);
  __hip_atomic_fetch_add(p + 2, v.z * w, __ATOMIC_RELAXED, __HIP_MEMORY_SCOPE_AGENT);
  __hip_atomic_fetch_add(p + 3, v.w * w, __ATOMIC_RELAXED, __HIP_MEMORY_SCOPE_AGENT);
}

// ---------------- Phase 2: out = agg @ W^T via V_WMMA_F32_16X16X4_F32 --------
// Block = 256 threads = 8 waves. Block handles one 16-node row tile; each wave
// produces one 16x16 output tile (wave w -> output cols [16w, 16w+16)).
// WMMA f32 16x16x4 fragment layout (ISA 7.12.2):
//   A 16x4:  lanes 0-15 hold (M=lane, K=0..1) in 2 VGPRs; lanes 16-31 K=2..3
//   B 4x16:  mirrored: lanes 0-15 hold (K=0..1, N=lane); lanes 16-31 K=2..3
//   C/D:     VGPR i: lanes 0-15 = (M=i, N=lane); lanes 16-31 = (M=i+8, N=lane-16)
__global__ __launch_bounds__(256) void gcn_gemm_kernel(
    const float* __restrict__ agg, const float* __restrict__ W,
    float* __restrict__ out) {
  __shared__ float atile[16 * LDS_STRIDE];

  const int tile = blockIdx.x;           // node tile: rows [16*tile, 16*tile+16)
  const int t = threadIdx.x;

  // Cooperative stage of the 16x128 A tile into padded LDS.
  const float* asrc = agg + (size_t)tile * 16 * D;
  for (int i = t; i < 16 * D; i += 256) {
    int r = i >> 7;      // / 128
    int c = i & 127;     // % 128
    atile[r * LDS_STRIDE + c] = asrc[i];
  }
  __syncthreads();

  const int wave = t >> 5;
  const int lane = t & 31;
  const int half = lane >> 4;   // selects K-pair {0,1} vs {2,3}
  const int l16 = lane & 15;
  const int obase = wave * 16;

  // B fragment source: B[k][n] = W[n][k]; W row-major -> contiguous float2.
  const float* wrow = W + (size_t)(obase + l16) * D + half * 2;
  const float* arow = atile + l16 * LDS_STRIDE + half * 2;

  v8f c = {};
#pragma unroll
  for (int k = 0; k < D; k += 4) {
    v2f a = *(const v2f*)(arow + k);   // ds_load_b64, conflict-free (pad 132)
    v2f b = *(const v2f*)(wrow + k);   // global float2 (W resident in L0/L2)
    c = __builtin_amdgcn_wmma_f32_16x16x4_f32(
        /*neg_a=*/false, a, /*neg_b=*/false, b,
        /*c_mod=*/(short)0, c, /*reuse_a=*/false, /*reuse_b=*/false);
  }

  // Scatter D fragment: row M = i + 8*half, col N = l16 within this tile.
  float* orow = out + ((size_t)tile * 16 + half * 8) * D + obase + l16;
#pragma unroll
  for (int i = 0; i < 8; ++i) {
    orow[(size_t)i * D] = c[i];
  }
}

extern "C" void kernel_launch(void* const* d_in, const int* in_sizes, int n_in,
                              void* d_out, int out_size, void* d_ws, size_t ws_size,
                              hipStream_t stream) {
  const float* node_emb    = (const float*)d_in[0];  // [N, 128]
  const float* edge_weight = (const float*)d_in[1];  // [E]
  const int*   src         = (const int*)d_in[2];    // [E]
  const int*   dst         = (const int*)d_in[3];    // [E]
  const float* W           = (const float*)d_in[4];  // [128, 128] row-major
  float* out = (float*)d_out;                        // [N, 128]

  const int N = in_sizes[0] / D;   // 50000
  const int E = in_sizes[1];       // 640000

  float* agg = (float*)d_ws;       // N*128 floats = 25.6 MB scratch

  // Phase 0: zero agg (deterministic per call; graph-capture safe).
  int total4 = (N * D) / 4;
  gcn_zero_kernel<<<(total4 + 255) / 256, 256, 0, stream>>>((float4*)agg, total4);

  // Phase 1: one wave per edge, 8 edges per 256-thread block.
  int nblocks = (E + 7) / 8;
  gcn_scatter_kernel<<<nblocks, 256, 0, stream>>>(node_emb, edge_weight, src, dst,
                                                  agg, E);

  // Phase 2: N/16 row tiles (50000 = 3125*16 exact -> EXEC all-ones for WMMA).
  gcn_gemm_kernel<<<N / 16, 256, 0, stream>>>(agg, W, out);
}